// TopKRouter_23965917511798
// MI455X (gfx1250) — compile-verified
//
#include <hip/hip_runtime.h>
#include <math.h>

typedef __attribute__((ext_vector_type(2))) float v2f;
typedef __attribute__((ext_vector_type(8))) float v8f;
typedef __attribute__((ext_vector_type(4))) int   v4i;

#define D_DIM 2048
#define E_DIM 64
#define KC    32
#define TOKS_PER_WG 128

// Per-buffer LDS layout (floats): Bs[64*36]=2304 | As: 8 waves * [16*36]=576 -> 6912 total.
// Epilogue logits reuse the front of the arena: 8 waves * [16*65] = 8320 floats.
#define BUF_FLOATS 6912

#if __has_builtin(__builtin_amdgcn_global_load_async_to_lds_b128) && \
    __has_builtin(__builtin_amdgcn_s_wait_asynccnt)
#define USE_ASYNC 1
#define SMEM_FLOATS (2 * BUF_FLOATS)          // 13824 (covers 8320 logits region)
#else
#define USE_ASYNC 0
#define SMEM_FLOATS 8320                      // max(6912 gemm, 8320 logits)
#endif

#if USE_ASYNC
// builtin signature (from hipcc diagnostic): (v4i addrspace(1)*, v4i addrspace(3)*, int, int)
#define GAS1(p) ((__attribute__((address_space(1))) v4i*)(p))
#define LAS3(p) ((__attribute__((address_space(3))) v4i*)(p))
#endif

__global__ __launch_bounds__(256) void router_init_ws(float* __restrict__ ws) {
    int t = threadIdx.x;
    if (t < 129) ws[t] = 0.0f;
}

__global__ __launch_bounds__(256) void router_main(
    const float* __restrict__ x, const float* __restrict__ gw,
    float* __restrict__ out_idx, float* __restrict__ out_w,
    float* __restrict__ ws, int tokens) {

    __shared__ float smem[SMEM_FLOATS];
    __shared__ float acc[129];   // [0..63]=softmax sums, [64..127]=counts, [128]=sum lse^2

    const int tid    = threadIdx.x;
    const int wave   = tid >> 5;
    const int lane   = tid & 31;
    const int lane16 = lane & 15;
    const int half   = lane >> 4;          // 0: K pair {0,1}, 1: K pair {2,3}

    if (tid < 129) acc[tid] = 0.0f;
    if ((long)blockIdx.x * TOKS_PER_WG >= tokens) return;   // block-uniform guard

    const long tokBase = (long)blockIdx.x * TOKS_PER_WG + wave * 16;
    const float* xw = x + (size_t)tokBase * D_DIM;

    // B staging ownership: thread t covers two 16B blocks (flat = t, t+256)
    //   e = flat>>3 (expert row), sub = flat&7 (16B block in 32-float row)
    // A staging ownership: lane covers token m = lane>>1, cols kb..kb+15 (4 x 16B)
    const int m_a  = lane >> 1;
    const int kb_a = (lane & 1) * 16;

    v8f c0 = {}, c1 = {}, c2 = {}, c3 = {};

#if USE_ASYNC
    // ---------------- async double-buffered pipeline ----------------
    int cur = 0;
    {   // prologue: stage chunk 0 into buffer 0
        float* Bs = smem;
        float* As = smem + 2304 + wave * 576;
        #pragma unroll
        for (int j = 0; j < 2; ++j) {
            int flat = tid + j * 256, e = flat >> 3, sub = flat & 7;
            __builtin_amdgcn_global_load_async_to_lds_b128(
                GAS1(gw + (size_t)e * D_DIM + sub * 4), LAS3(Bs + e * 36 + sub * 4), 0, 0);
        }
        #pragma unroll
        for (int q = 0; q < 4; ++q) {
            __builtin_amdgcn_global_load_async_to_lds_b128(
                GAS1(xw + (size_t)m_a * D_DIM + kb_a + q * 4),
                LAS3(As + m_a * 36 + kb_a + q * 4), 0, 0);
        }
        __builtin_amdgcn_s_wait_asynccnt(0);
        __syncthreads();
    }
    for (int kc = 0; kc < D_DIM; kc += KC) {
        if (kc + KC < D_DIM) {      // stage next chunk into the other buffer
            float* Bn = smem + (1 - cur) * BUF_FLOATS;
            float* An = Bn + 2304 + wave * 576;
            #pragma unroll
            for (int j = 0; j < 2; ++j) {
                int flat = tid + j * 256, e = flat >> 3, sub = flat & 7;
                __builtin_amdgcn_global_load_async_to_lds_b128(
                    GAS1(gw + (size_t)e * D_DIM + kc + KC + sub * 4),
                    LAS3(Bn + e * 36 + sub * 4), 0, 0);
            }
            #pragma unroll
            for (int q = 0; q < 4; ++q) {
                __builtin_amdgcn_global_load_async_to_lds_b128(
                    GAS1(xw + (size_t)m_a * D_DIM + kc + KC + kb_a + q * 4),
                    LAS3(An + m_a * 36 + kb_a + q * 4), 0, 0);
            }
        }
        const float* Bs = smem + cur * BUF_FLOATS;
        const float* As = Bs + 2304 + wave * 576;
        #pragma unroll
        for (int kk = 0; kk < KC; kk += 4) {
            const int kA = kk + half * 2;
            v2f a;  a.x = As[lane16 * 36 + kA];  a.y = As[lane16 * 36 + kA + 1];
            v2f b0, b1, b2, b3;
            b0.x = Bs[(lane16 +  0) * 36 + kA]; b0.y = Bs[(lane16 +  0) * 36 + kA + 1];
            b1.x = Bs[(lane16 + 16) * 36 + kA]; b1.y = Bs[(lane16 + 16) * 36 + kA + 1];
            b2.x = Bs[(lane16 + 32) * 36 + kA]; b2.y = Bs[(lane16 + 32) * 36 + kA + 1];
            b3.x = Bs[(lane16 + 48) * 36 + kA]; b3.y = Bs[(lane16 + 48) * 36 + kA + 1];
            c0 = __builtin_amdgcn_wmma_f32_16x16x4_f32(false, a, false, b0, (short)0, c0, false, false);
            c1 = __builtin_amdgcn_wmma_f32_16x16x4_f32(false, a, false, b1, (short)0, c1, false, false);
            c2 = __builtin_amdgcn_wmma_f32_16x16x4_f32(false, a, false, b2, (short)0, c2, false, false);
            c3 = __builtin_amdgcn_wmma_f32_16x16x4_f32(false, a, false, b3, (short)0, c3, false, false);
        }
        __builtin_amdgcn_s_wait_asynccnt(0);
        __syncthreads();
        cur ^= 1;
    }
#else
    // ---------------- register double-buffered pipeline ----------------
    float* Bs = smem;                       // [e*36 + sub*4 ...]
    float* As = smem + 2304 + wave * 576;   // [m*36 + k]
    float4 bReg[2], aReg[4];
    {   // prologue: load chunk 0 into registers
        #pragma unroll
        for (int j = 0; j < 2; ++j) {
            int flat = tid + j * 256, e = flat >> 3, sub = flat & 7;
            bReg[j] = *(const float4*)(gw + (size_t)e * D_DIM + sub * 4);
        }
        #pragma unroll
        for (int q = 0; q < 4; ++q)
            aReg[q] = *(const float4*)(xw + (size_t)m_a * D_DIM + kb_a + q * 4);
    }
    for (int kc = 0; kc < D_DIM; kc += KC) {
        // commit staged registers to LDS
        #pragma unroll
        for (int j = 0; j < 2; ++j) {
            int flat = tid + j * 256, e = flat >> 3, sub = flat & 7;
            *(float4*)(Bs + e * 36 + sub * 4) = bReg[j];
        }
        #pragma unroll
        for (int q = 0; q < 4; ++q)
            *(float4*)(As + m_a * 36 + kb_a + q * 4) = aReg[q];
        __syncthreads();
        // issue next chunk's global loads; latency hidden behind WMMA below
        if (kc + KC < D_DIM) {
            #pragma unroll
            for (int j = 0; j < 2; ++j) {
                int flat = tid + j * 256, e = flat >> 3, sub = flat & 7;
                bReg[j] = *(const float4*)(gw + (size_t)e * D_DIM + kc + KC + sub * 4);
            }
            #pragma unroll
            for (int q = 0; q < 4; ++q)
                aReg[q] = *(const float4*)(xw + (size_t)m_a * D_DIM + kc + KC + kb_a + q * 4);
        }
        #pragma unroll
        for (int kk = 0; kk < KC; kk += 4) {
            const int kA = kk + half * 2;
            v2f a;  a.x = As[lane16 * 36 + kA];  a.y = As[lane16 * 36 + kA + 1];
            v2f b0, b1, b2, b3;
            b0.x = Bs[(lane16 +  0) * 36 + kA]; b0.y = Bs[(lane16 +  0) * 36 + kA + 1];
            b1.x = Bs[(lane16 + 16) * 36 + kA]; b1.y = Bs[(lane16 + 16) * 36 + kA + 1];
            b2.x = Bs[(lane16 + 32) * 36 + kA]; b2.y = Bs[(lane16 + 32) * 36 + kA + 1];
            b3.x = Bs[(lane16 + 48) * 36 + kA]; b3.y = Bs[(lane16 + 48) * 36 + kA + 1];
            c0 = __builtin_amdgcn_wmma_f32_16x16x4_f32(false, a, false, b0, (short)0, c0, false, false);
            c1 = __builtin_amdgcn_wmma_f32_16x16x4_f32(false, a, false, b1, (short)0, c1, false, false);
            c2 = __builtin_amdgcn_wmma_f32_16x16x4_f32(false, a, false, b2, (short)0, c2, false, false);
            c3 = __builtin_amdgcn_wmma_f32_16x16x4_f32(false, a, false, b3, (short)0, c3, false, false);
        }
        __syncthreads();
    }
#endif

    // ---- dump 16x64 logits tile to LDS (stride 65 -> conflict-free row scans)
    float* Ls = smem + wave * 1040;
    #pragma unroll
    for (int r = 0; r < 8; ++r) {
        int row = half ? (r + 8) : r;
        Ls[row * 65 +  0 + lane16] = c0[r];
        Ls[row * 65 + 16 + lane16] = c1[r];
        Ls[row * 65 + 32 + lane16] = c2[r];
        Ls[row * 65 + 48 + lane16] = c3[r];
    }
    __syncthreads();

    // ---- per-token softmax + top-2 + loss partials (lanes 0..15 each own a token)
    if (lane < 16) {
        const float* lr = Ls + lane * 65;
        float mx = -3.0e38f;
        #pragma unroll
        for (int e = 0; e < E_DIM; ++e) mx = fmaxf(mx, lr[e]);
        float sum = 0.0f;
        float l1 = -3.0e38f, l2 = -3.0e38f;
        int i1 = 0, i2 = 0;
        #pragma unroll
        for (int e = 0; e < E_DIM; ++e) {
            float v = lr[e];
            sum += __expf(v - mx);
            if (v > l1)      { l2 = l1; i2 = i1; l1 = v; i1 = e; }
            else if (v > l2) { l2 = v;  i2 = e; }
        }
        float lse = mx + __logf(sum);
        #pragma unroll
        for (int e = 0; e < E_DIM; ++e) atomicAdd(&acc[e], __expf(lr[e] - lse));
        atomicAdd(&acc[64 + i1], 1.0f);
        atomicAdd(&acc[64 + i2], 1.0f);
        atomicAdd(&acc[128], lse * lse);

        float w1 = __expf(l1 - lse), w2 = __expf(l2 - lse);
        float inv = 1.0f / (w1 + w2);
        long tok = tokBase + lane;
        out_idx[tok * 2 + 0] = (float)i1;
        out_idx[tok * 2 + 1] = (float)i2;
        out_w[tok * 2 + 0] = w1 * inv;
        out_w[tok * 2 + 1] = w2 * inv;
    }
    __syncthreads();

    if (tid < 129) atomicAdd(&ws[tid], acc[tid]);
}

__global__ __launch_bounds__(64) void router_finish(
    const float* __restrict__ ws, float* __restrict__ out_aux, float inv_tokens) {
    __shared__ float red[64];
    int t = threadIdx.x;
    float f = ws[64 + t] * inv_tokens * 0.5f;  // counts / (tokens * K)
    float P = ws[t] * inv_tokens;              // mean softmax weight
    red[t] = f * P;
    __syncthreads();
    #pragma unroll
    for (int s = 32; s > 0; s >>= 1) {
        if (t < s) red[t] += red[t + s];
        __syncthreads();
    }
    if (t == 0) {
        float balance = (float)E_DIM * red[0];
        float z = ws[128] * inv_tokens;
        out_aux[0] = 0.01f * balance + 0.001f * z;
    }
}

extern "C" void kernel_launch(void* const* d_in, const int* in_sizes, int n_in,
                              void* d_out, int out_size, void* d_ws, size_t ws_size,
                              hipStream_t stream) {
    const float* x  = (const float*)d_in[0];
    const float* gw = (const float*)d_in[1];
    const int tokens = in_sizes[0] / D_DIM;          // B*S = 16384

    float* ws      = (float*)d_ws;
    float* out     = (float*)d_out;
    float* out_idx = out;                             // tokens*2 floats (indices as float)
    float* out_w   = out + (size_t)tokens * 2;        // tokens*2 floats
    float* out_aux = out + (size_t)tokens * 4;        // 1 float

    router_init_ws<<<1, 256, 0, stream>>>(ws);
    const int blocks = (tokens + TOKS_PER_WG - 1) / TOKS_PER_WG;
    router_main<<<blocks, 256, 0, stream>>>(x, gw, out_idx, out_w, ws, tokens);
    router_finish<<<1, 64, 0, stream>>>(ws, out_aux, 1.0f / (float)tokens);
}